// Block1_43542378447225
// MI455X (gfx1250) — compile-verified
//
#include <hip/hip_runtime.h>
#include <math.h>

// ---------------------------------------------------------------------------
// MI455X (gfx1250) implementation. wave32. f32 WMMA (V_WMMA_F32_16X16X4_F32)
// for all GEMMs; Jacobian never materialized (piecewise-linear analytic form).
// ---------------------------------------------------------------------------

typedef float v2f __attribute__((ext_vector_type(2)));
typedef float v8f __attribute__((ext_vector_type(8)));

__device__ __forceinline__ v8f wmma4(v2f a, v2f b, v8f c) {
  // 8-arg pattern: (neg_a, A, neg_b, B, c_mod, C, reuse_a, reuse_b)
  return __builtin_amdgcn_wmma_f32_16x16x4_f32(false, a, false, b, (short)0, c,
                                               false, false);
}

// workspace layout (float offsets)
#define WS_KV    0        // 512*64      K @ Vw
#define WS_ZF    32768    // 128*64      z2 in (row=b*64+pq, c) layout
#define WS_M2    40960    // 128*64      relu' mask of conv2 pre-activation
#define WS_Y1    49152    // 2*32*16*16  y1 = relu(conv1)
#define WS_D     65536    // 128*64      D = 2*(z2 - q)
#define WS_ES    73728    // 128*1024    e_sum (b,pq,i,j)
#define WS_EMIN  204800   // 2*1024      min over pq
#define WS_ZFM   206848   // 128*64      z2_masked rows
// total 215040 floats = 860160 bytes

// ---------------------------------------------------------------------------
// KV = K(512x64) @ Vw(64x64).  32 blocks x 4 waves; one 16x16 tile per wave.
// ---------------------------------------------------------------------------
__global__ void kv_gemm(const float* __restrict__ K, const float* __restrict__ Vw,
                        float* __restrict__ KV) {
  const int lane = threadIdx.x & 31;
  const int wave = threadIdx.x >> 5;   // N-tile 0..3
  const int mt   = blockIdx.x;         // M-tile 0..31
  const int row  = lane & 15;
  const int hi   = lane >> 4;
  const int koff = hi * 2;
  const int col  = row;
  v8f acc = {};
  const float* Arow = K + (mt * 16 + row) * 64;
  for (int k0 = 0; k0 < 64; k0 += 4) {
    v2f a = *(const v2f*)(Arow + k0 + koff);
    v2f b;
    b[0] = Vw[(k0 + koff    ) * 64 + wave * 16 + col];
    b[1] = Vw[(k0 + koff + 1) * 64 + wave * 16 + col];
    acc = wmma4(a, b, acc);
  }
#pragma unroll
  for (int r = 0; r < 8; ++r)
    KV[(mt * 16 + r + hi * 8) * 64 + wave * 16 + col] = acc[r];
}

// ---------------------------------------------------------------------------
// Fused forward: conv1(4x4,s2,p1)+relu -> LDS, conv2(4x4,s2,p1)+relu.
// Stores y1, zf (row-major rows = b*64+pq), relu' mask of conv2.
// ---------------------------------------------------------------------------
__global__ void fwd_convs(const float* __restrict__ x, const float* __restrict__ w1,
                          const float* __restrict__ b1, const float* __restrict__ w2,
                          const float* __restrict__ b2, float* __restrict__ y1w,
                          float* __restrict__ zf, float* __restrict__ m2w) {
  __shared__ float y1s[32 * 16 * 16];
  const int b = blockIdx.x;
  const int t = threadIdx.x;  // 256 threads
  for (int it = 0; it < 32; ++it) {
    int idx = t + it * 256;                       // 8192 y1 elements
    int m = idx >> 8, rem = idx & 255, u = rem >> 4, v = rem & 15;
    float pre = b1[m];
    for (int h = 0; h < 3; ++h)
      for (int di = 0; di < 4; ++di) {
        int i = 2 * u - 1 + di;
        if (i < 0 || i >= 32) continue;
        for (int dj = 0; dj < 4; ++dj) {
          int j = 2 * v - 1 + dj;
          if (j < 0 || j >= 32) continue;
          pre += w1[((m * 3 + h) * 4 + di) * 4 + dj] *
                 x[((b * 3 + h) * 32 + i) * 32 + j];
        }
      }
    float val = fmaxf(pre, 0.f);
    y1s[idx] = val;
    y1w[b * 8192 + idx] = val;
  }
  __syncthreads();
  for (int it = 0; it < 16; ++it) {
    int idx = t + it * 256;                       // 4096 z2 elements
    int c = idx >> 6, rem = idx & 63, p = rem >> 3, q = rem & 7;
    float pre = b2[c];
    for (int m = 0; m < 32; ++m)
      for (int du = 0; du < 4; ++du) {
        int u = 2 * p - 1 + du;
        if (u < 0 || u >= 16) continue;
        for (int dv = 0; dv < 4; ++dv) {
          int v = 2 * q - 1 + dv;
          if (v < 0 || v >= 16) continue;
          pre += w2[((c * 32 + m) * 4 + du) * 4 + dv] * y1s[(m * 16 + u) * 16 + v];
        }
      }
    int rowg = b * 64 + rem;
    zf[rowg * 64 + c]  = fmaxf(pre, 0.f);
    m2w[rowg * 64 + c] = (pre > 0.f) ? 1.f : 0.f;
  }
}

// ---------------------------------------------------------------------------
// Hopfield lookup on 16 rows per block (8 blocks x 8 waves).
// Phase1: logits(16x512) = beta * zin @ K^T  (WMMA, 4 N-tiles per wave)
// Phase2: row-wise softmax in LDS
// Phase3: q(16x64) = attn @ KV               (WMMA, waves 0..3)
// mode 0: out = D = 2*(zin - q)  (row-major)      mode 1: out = q as (b,c,p,q)
// ---------------------------------------------------------------------------
#define LSTR 520
__global__ void hopfield(const float* __restrict__ zin, const float* __restrict__ Kmat,
                         const float* __restrict__ KV, float* __restrict__ out,
                         int mode) {
  __shared__ float lg[16 * LSTR];
  __shared__ float red[16 * 16];
  __shared__ float rowred[16];
  const int lane = threadIdx.x & 31;
  const int wave = threadIdx.x >> 5;   // 0..7
  const int rt   = blockIdx.x;         // 0..7 -> rows rt*16..+15
  const int row  = lane & 15;
  const int hi   = lane >> 4;
  const int koff = hi * 2;
  const int col  = row;
  const float beta = 0.125f;           // 1/sqrt(64)

  // phase 1
  for (int nt = wave; nt < 32; nt += 8) {
    v8f acc = {};
    const float* Arow = zin + (rt * 16 + row) * 64;
    const float* Brow = Kmat + (nt * 16 + col) * 64;   // B[k,n] = K[n,k]
    for (int k0 = 0; k0 < 64; k0 += 4) {
      v2f a = *(const v2f*)(Arow + k0 + koff);
      v2f b = *(const v2f*)(Brow + k0 + koff);
      acc = wmma4(a, b, acc);
    }
#pragma unroll
    for (int r = 0; r < 8; ++r)
      lg[(r + hi * 8) * LSTR + nt * 16 + col] = beta * acc[r];
  }
  __syncthreads();

  // phase 2: softmax over 512 per row; 16 threads per row, 32 cols each
  {
    const int r = threadIdx.x >> 4;
    const int seg = threadIdx.x & 15;
    float* Lr = lg + r * LSTR + seg * 32;
    float mx = -1e30f;
    for (int c = 0; c < 32; ++c) mx = fmaxf(mx, Lr[c]);
    red[r * 16 + seg] = mx;
    __syncthreads();
    if (seg == 0) {
      float m2 = red[r * 16];
      for (int c2 = 1; c2 < 16; ++c2) m2 = fmaxf(m2, red[r * 16 + c2]);
      rowred[r] = m2;
    }
    __syncthreads();
    float rmx = rowred[r];
    float s = 0.f;
    for (int c = 0; c < 32; ++c) { float e = expf(Lr[c] - rmx); Lr[c] = e; s += e; }
    __syncthreads();
    red[r * 16 + seg] = s;
    __syncthreads();
    if (seg == 0) {
      float s2 = 0.f;
      for (int c2 = 0; c2 < 16; ++c2) s2 += red[r * 16 + c2];
      rowred[r] = 1.f / s2;
    }
    __syncthreads();
    float inv = rowred[r];
    for (int c = 0; c < 32; ++c) Lr[c] *= inv;
  }
  __syncthreads();

  // phase 3: q = attn(16x512) @ KV(512x64); waves 0..3 (uniform branch, EXEC all-1)
  if (wave < 4) {
    const int nt = wave;
    v8f acc = {};
    const float* Arow = lg + row * LSTR;
    for (int k0 = 0; k0 < 512; k0 += 4) {
      v2f a = *(const v2f*)(Arow + k0 + koff);
      v2f b;
      b[0] = KV[(k0 + koff    ) * 64 + nt * 16 + col];
      b[1] = KV[(k0 + koff + 1) * 64 + nt * 16 + col];
      acc = wmma4(a, b, acc);
    }
#pragma unroll
    for (int r = 0; r < 8; ++r) {
      int grow = rt * 16 + r + hi * 8;
      int gcol = nt * 16 + col;
      float qv = acc[r];
      if (mode == 0) {
        out[grow * 64 + gcol] = 2.f * (zin[grow * 64 + gcol] - qv);
      } else {
        int bb = grow >> 6, pq = grow & 63;
        out[bb * 4096 + gcol * 64 + pq] = qv;   // (b,c,p,q)
      }
    }
  }
}

// ---------------------------------------------------------------------------
// e_sum[b,p,q,:,:] = backprop of (D*mask2)[b,:,p,q] through W2,mask1,W1,
// summed over input channels. One block per (b,p,q), 64 threads.
// ---------------------------------------------------------------------------
__global__ void esum_k(const float* __restrict__ Dw, const float* __restrict__ m2w,
                       const float* __restrict__ y1w, const float* __restrict__ w1,
                       const float* __restrict__ w2, float* __restrict__ es) {
  __shared__ float delta2[64];
  __shared__ float g1[512];    // [m][du][dv]
  __shared__ float w1s[512];   // sum_h w1[m,h,di,dj]
  const int rowg = blockIdx.x;
  const int b = rowg >> 6, pq = rowg & 63, p = pq >> 3, q = pq & 7;
  const int t = threadIdx.x;
  delta2[t] = Dw[rowg * 64 + t] * m2w[rowg * 64 + t];
  for (int l = t; l < 512; l += 64) {
    int m = l >> 4, di = (l >> 2) & 3, dj = l & 3;
    w1s[l] = w1[((m * 3 + 0) * 4 + di) * 4 + dj] +
             w1[((m * 3 + 1) * 4 + di) * 4 + dj] +
             w1[((m * 3 + 2) * 4 + di) * 4 + dj];
  }
  __syncthreads();
  for (int l = t; l < 512; l += 64) {
    int m = l >> 4, du = (l >> 2) & 3, dv = l & 3;
    int u = 2 * p - 1 + du, v = 2 * q - 1 + dv;
    float s = 0.f;
    if (u >= 0 && u < 16 && v >= 0 && v < 16 &&
        y1w[b * 8192 + (m * 16 + u) * 16 + v] > 0.f) {
      for (int c = 0; c < 64; ++c)
        s += delta2[c] * w2[((c * 32 + m) * 4 + du) * 4 + dv];
    }
    g1[l] = s;
  }
  __syncthreads();
  for (int l = t; l < 1024; l += 64) {           // full 32x32 map, zeros outside RF
    int i = l >> 5, j = l & 31;
    int oi = i - (4 * p - 3), oj = j - (4 * q - 3);
    float val = 0.f;
    if (oi >= 0 && oi < 10 && oj >= 0 && oj < 10) {
      for (int du = 0; du < 4; ++du) {
        int di = oi - 2 * du;
        if (di < 0 || di > 3) continue;
        for (int dv = 0; dv < 4; ++dv) {
          int dj = oj - 2 * dv;
          if (dj < 0 || dj > 3) continue;
          for (int m = 0; m < 32; ++m)
            val += g1[m * 16 + du * 4 + dv] * w1s[m * 16 + di * 4 + dj];
        }
      }
    }
    es[rowg * 1024 + l] = val;
  }
}

// min over the 64 (p,q) planes per (b,i,j)
__global__ void emin_k(const float* __restrict__ es, float* __restrict__ em) {
  int gid = blockIdx.x * 256 + threadIdx.x;   // 0..2047
  int b = gid >> 10, ij = gid & 1023;
  const float* base = es + b * 65536 + ij;
  float mn = base[0];
  for (int pq = 1; pq < 64; ++pq) mn = fminf(mn, base[pq * 1024]);
  em[gid] = mn;
}

// ---------------------------------------------------------------------------
// z2_masked[b,:,p,q]: forward pass of masked x (10x10 RF) through the
// frozen-mask linearization. One block per (b,p,q), 64 threads.
// ---------------------------------------------------------------------------
__global__ void zmask_k(const float* __restrict__ x, const float* __restrict__ w1,
                        const float* __restrict__ w2, const float* __restrict__ y1w,
                        const float* __restrict__ m2w, const float* __restrict__ es,
                        const float* __restrict__ em, float* __restrict__ zfm) {
  __shared__ float xm[3 * 10 * 10];
  __shared__ float y1m[512];
  const int rowg = blockIdx.x;
  const int b = rowg >> 6, pq = rowg & 63, p = pq >> 3, q = pq & 7;
  const int t = threadIdx.x;
  for (int l = t; l < 300; l += 64) {
    int h = l / 100, rr = l % 100, oi = rr / 10, oj = rr % 10;
    int i = 4 * p - 3 + oi, j = 4 * q - 3 + oj;
    float val = 0.f;
    if (i >= 0 && i < 32 && j >= 0 && j < 32) {
      if (es[rowg * 1024 + i * 32 + j] <= em[b * 1024 + i * 32 + j])  // mask==1
        val = x[((b * 3 + h) * 32 + i) * 32 + j];
    }
    xm[l] = val;
  }
  __syncthreads();
  for (int l = t; l < 512; l += 64) {
    int m = l >> 4, du = (l >> 2) & 3, dv = l & 3;
    int u = 2 * p - 1 + du, v = 2 * q - 1 + dv;
    float s = 0.f;
    if (u >= 0 && u < 16 && v >= 0 && v < 16 &&
        y1w[b * 8192 + (m * 16 + u) * 16 + v] > 0.f) {
      for (int h = 0; h < 3; ++h)
        for (int di = 0; di < 4; ++di)
          for (int dj = 0; dj < 4; ++dj)
            s += w1[((m * 3 + h) * 4 + di) * 4 + dj] *
                 xm[h * 100 + (2 * du + di) * 10 + (2 * dv + dj)];
    }
    y1m[l] = s;
  }
  __syncthreads();
  {
    int c = t;
    float s = 0.f;
    for (int m = 0; m < 32; ++m)
      for (int du = 0; du < 4; ++du)
        for (int dv = 0; dv < 4; ++dv)
          s += w2[((c * 32 + m) * 4 + du) * 4 + dv] * y1m[m * 16 + du * 4 + dv];
    zfm[rowg * 64 + c] = s * m2w[rowg * 64 + c];
  }
}

// ---------------------------------------------------------------------------
extern "C" void kernel_launch(void* const* d_in, const int* in_sizes, int n_in,
                              void* d_out, int out_size, void* d_ws, size_t ws_size,
                              hipStream_t stream) {
  (void)in_sizes; (void)n_in; (void)out_size; (void)ws_size;
  const float* x  = (const float*)d_in[0];
  const float* w1 = (const float*)d_in[1];
  const float* b1 = (const float*)d_in[2];
  const float* w2 = (const float*)d_in[3];
  const float* b2 = (const float*)d_in[4];
  const float* Km = (const float*)d_in[5];
  const float* Vw = (const float*)d_in[6];
  float* out = (float*)d_out;
  float* ws  = (float*)d_ws;

  float* KV  = ws + WS_KV;
  float* zf  = ws + WS_ZF;
  float* m2w = ws + WS_M2;
  float* y1w = ws + WS_Y1;
  float* Dw  = ws + WS_D;
  float* es  = ws + WS_ES;
  float* em  = ws + WS_EMIN;
  float* zfm = ws + WS_ZFM;

  kv_gemm  <<< 32, 128, 0, stream>>>(Km, Vw, KV);
  fwd_convs<<<  2, 256, 0, stream>>>(x, w1, b1, w2, b2, y1w, zf, m2w);
  hopfield <<<  8, 256, 0, stream>>>(zf, Km, KV, Dw, 0);
  esum_k   <<<128,  64, 0, stream>>>(Dw, m2w, y1w, w1, w2, es);
  emin_k   <<<  8, 256, 0, stream>>>(es, em);
  zmask_k  <<<128,  64, 0, stream>>>(x, w1, w2, y1w, m2w, es, em, zfm);
  hopfield <<<  8, 256, 0, stream>>>(zfm, Km, KV, out, 1);
}